// XConv_52218212385432
// MI455X (gfx1250) — compile-verified
//
#include <hip/hip_runtime.h>
#include <math.h>

typedef __attribute__((ext_vector_type(2))) float v2f;
typedef __attribute__((ext_vector_type(8))) float v8f;
typedef __attribute__((ext_vector_type(4))) unsigned int v4u;
typedef __attribute__((ext_vector_type(4))) int v4i;
typedef __attribute__((ext_vector_type(8))) int v8i;

#ifndef __has_builtin
#define __has_builtin(x) 0
#endif
#if __has_builtin(__builtin_amdgcn_tensor_load_to_lds)
#define HAVE_TDM 1
#else
#define HAVE_TDM 0
#endif
#if defined(__has_include)
#if __has_include(<hip/amd_detail/amd_gfx1250_TDM.h>)
#define TDM_SIX_ARGS 1
#else
#define TDM_SIX_ARGS 0
#endif
#else
#define TDM_SIX_ARGS 0
#endif

#define B_    8
#define CIN   64
#define NPTS  16384
#define QPTS  1024
#define KNN   16
#define COUT  128
#define MID   32
#define K2    256
#define CCAT  96
#define DMUL  2
#define EPS_  1e-5f

__device__ __forceinline__ float elu1(float v) { return v > 0.f ? v : expm1f(v); }

// ---------------- workspace layout (float elements) ----------------
#define OFF_QI    ((size_t)0)                       // B*Q ints          = 8192
#define OFF_KNN   ((size_t)8192)                    // B*K*Q ints        = 131072
#define OFF_PHAT  ((size_t)139264)                  // B*3*K*Q floats    = 393216
#define OFF_T1    ((size_t)532480)                  // B*256*Q           = 2097152
#define OFF_T2    ((size_t)2629632)                 // B*256*Q
#define OFF_TT    ((size_t)4726784)                 // B*256*Q
#define OFF_F2    ((size_t)6823936)                 // B*32*K*Q          = 4194304
#define OFF_DW    ((size_t)11018240)                // B*192*Q           = 1572864
// total = 12591104 floats ~= 50.4 MB

// ================= 1) Farthest point sampling =================
// One block per batch. dists[] lives in LDS (64KB of the 320KB WGP pool).
// Matches lax.scan semantics: emit incoming `far`, then update+argmax.
__global__ void fps_kernel(const float* __restrict__ x, int* __restrict__ qi)
{
    extern __shared__ float smem[];
    float* dists = smem;                 // NPTS
    float* redV  = smem + NPTS;          // 1024
    int*   redI  = (int*)(redV + 1024);  // 1024
    const int b   = blockIdx.x;
    const int tid = threadIdx.x;
    const float* px = x + (size_t)b * CIN * NPTS;
    const float* py = px + NPTS;
    const float* pz = px + 2 * NPTS;
    for (int j = tid; j < NPTS; j += 1024) dists[j] = 3.0e38f;
    __syncthreads();
    int far = 0;
    for (int i = 0; i < QPTS; ++i) {
        if (tid == 0) qi[b * QPTS + i] = far;
        float sx = px[far], sy = py[far], sz = pz[far];
        float bestV = -1.f; int bestI = 0;
        for (int j = tid; j < NPTS; j += 1024) {
            float dx = px[j] - sx, dy = py[j] - sy, dz = pz[j] - sz;
            float d = fminf(dists[j], dx*dx + dy*dy + dz*dz);
            dists[j] = d;
            if (d > bestV) { bestV = d; bestI = j; }   // strict > : lowest index wins ties
        }
        redV[tid] = bestV; redI[tid] = bestI;
        __syncthreads();
        for (int s = 512; s > 0; s >>= 1) {
            if (tid < s) {
                float v2 = redV[tid + s]; int i2 = redI[tid + s];
                if (v2 > redV[tid] || (v2 == redV[tid] && i2 < redI[tid])) {
                    redV[tid] = v2; redI[tid] = i2;
                }
            }
            __syncthreads();
        }
        far = redI[0];
        __syncthreads();
    }
}

// ================= 2) kNN (top-32 smallest, keep every 2nd) =================
// Point tiles (3 rows x 512 cols of x) are streamed into LDS by the Tensor
// Data Mover (double-buffered, TENSORcnt-tracked) while all waves scan the
// previous tile from LDS.
#define KTILE  512
#define NTILES (NPTS / KTILE)   // 32

#if HAVE_TDM
__device__ __forceinline__ void tdm_issue_tile(float* ldsPtr, const float* gptr)
{
    // D# group 0: count=1 | lds_addr | global_addr(57b) | type=2
    unsigned lds_off = (unsigned)(unsigned long long)(uintptr_t)ldsPtr; // low 32b of LDS addr
    unsigned long long ga = (unsigned long long)(uintptr_t)gptr;
    v4u g0 = { 1u,
               lds_off,
               (unsigned)(ga & 0xFFFFFFFFu),
               (unsigned)((ga >> 32) & 0x1FFFFFFu) | (2u << 30) };
    // D# group 1: data_size=4B; tensor 16384x3 (f32), stride0=16384; tile 512x3
    v8i g1 = { (int)(2u << 16),                       // data_size=2 (4 bytes)
               (int)((unsigned)(NPTS & 0xFFFF) << 16),// tensor_dim0[15:0] @ [63:48]
               (int)(3u << 16),                       // dim0[31:16]=0 | tensor_dim1=3
               (int)((unsigned)KTILE << 16),          // dim1[31:16]=0 | tile_dim0=512
               3,                                     // tile_dim1=3 | tile_dim2=0
               NPTS,                                  // tensor_dim0_stride[31:0]
               0,                                     // stride0 hi | stride1 lo
               0 };
    v4i z4 = { 0, 0, 0, 0 };
#if TDM_SIX_ARGS
    v8i z8 = { 0, 0, 0, 0, 0, 0, 0, 0 };
    __builtin_amdgcn_tensor_load_to_lds(g0, g1, z4, z4, z8, 0);
#else
    __builtin_amdgcn_tensor_load_to_lds(g0, g1, z4, z4, 0);
#endif
}
#endif

__global__ void knn_kernel(const float* __restrict__ x, const int* __restrict__ qi,
                           int* __restrict__ knn)
{
    __shared__ float stage[2][3 * KTILE];
    const int b = blockIdx.x / (QPTS / 256);
    const int q = (blockIdx.x % (QPTS / 256)) * 256 + threadIdx.x;
    const float* px = x + (size_t)b * CIN * NPTS;
    const float* py = px + NPTS;
    const float* pz = px + 2 * NPTS;
    const int qg = qi[b * QPTS + q];
    const float qx = px[qg], qy = py[qg], qz = pz[qg];
    const float qn = qx*qx + qy*qy + qz*qz;
    float bd[32]; int bi[32];
    #pragma unroll
    for (int i = 0; i < 32; ++i) { bd[i] = 3.0e38f; bi[i] = 0; }

#if HAVE_TDM
    if (threadIdx.x < 32) tdm_issue_tile(&stage[0][0], px);   // prime tile 0
#endif
    for (int t = 0; t < NTILES; ++t) {
        const int buf = t & 1;
#if HAVE_TDM
        if (threadIdx.x < 32) {
            if (t + 1 < NTILES) {
                tdm_issue_tile(&stage[(t + 1) & 1][0], px + (size_t)(t + 1) * KTILE);
                __builtin_amdgcn_s_wait_tensorcnt(1);  // tile t complete, t+1 in flight
            } else {
                __builtin_amdgcn_s_wait_tensorcnt(0);  // drain
            }
        }
        __syncthreads();
#else
        __syncthreads();
        for (int i = threadIdx.x; i < KTILE; i += 256) {
            stage[buf][0 * KTILE + i] = px[t * KTILE + i];
            stage[buf][1 * KTILE + i] = py[t * KTILE + i];
            stage[buf][2 * KTILE + i] = pz[t * KTILE + i];
        }
        __syncthreads();
#endif
        const float* sx = &stage[buf][0];
        const float* sy = sx + KTILE;
        const float* sz = sy + KTILE;
        const int t0 = t * KTILE;
        for (int i = 0; i < KTILE; ++i) {
            float ax = sx[i], ay = sy[i], az = sz[i];
            // expanded-form distance exactly as the reference computes it
            float d = qn + (ax*ax + ay*ay + az*az) - 2.f * (qx*ax + qy*ay + qz*az);
            if (d < bd[31]) {
                int pos = 31;
                while (pos > 0 && bd[pos - 1] > d) {
                    bd[pos] = bd[pos - 1]; bi[pos] = bi[pos - 1]; --pos;
                }
                bd[pos] = d; bi[pos] = t0 + i;
            }
        }
        __syncthreads();   // scan done before this buffer is DMA'd again
    }
    for (int k = 0; k < KNN; ++k)
        knn[(size_t)b * KNN * QPTS + (size_t)k * QPTS + q] = bi[2 * k];   // ::DIL
}

// ================= 3) p_hat + emit query coords into d_out =================
__global__ void phat_kernel(const float* __restrict__ x, const int* __restrict__ qi,
                            const int* __restrict__ knn, float* __restrict__ phat,
                            float* __restrict__ out)
{
    int i = blockIdx.x * 256 + threadIdx.x;      // B*3*K*Q
    int q = i & (QPTS - 1);
    int k = (i >> 10) & (KNN - 1);
    int c = (i >> 14) % 3;
    int b = i / (3 * KNN * QPTS);
    const float* xb = x + (size_t)b * CIN * NPTS + (size_t)c * NPTS;
    int qg = qi[b * QPTS + q];
    int ng = knn[(size_t)b * KNN * QPTS + (size_t)k * QPTS + q];
    float qc = xb[qg];
    phat[(((size_t)b * 3 + c) * KNN + k) * QPTS + q] = xb[ng] - qc;
    if (k == 0) out[((size_t)b * 131 + c) * QPTS + q] = qc;
}

// ================= 4) t1 = elu(bn(cK1(p_hat, sw1) + sb1)) =================
__global__ void t1_kernel(const float* __restrict__ phat, const float* __restrict__ sw1,
                          const float* __restrict__ sb1,
                          const float* __restrict__ g, const float* __restrict__ bb,
                          const float* __restrict__ rm, const float* __restrict__ rv,
                          float* __restrict__ t1)
{
    int i = blockIdx.x * 256 + threadIdx.x;      // B*256*Q
    int q = i & (QPTS - 1);
    int o = (i >> 10) & 255;
    int b = i >> 18;
    const float* ph = phat + (size_t)b * 3 * KNN * QPTS + q;
    const float* w  = sw1 + (size_t)o * 48;
    float acc = sb1[o];
    #pragma unroll
    for (int ck = 0; ck < 48; ++ck) acc += w[ck] * ph[(size_t)ck * QPTS];
    float s = g[o] * rsqrtf(rv[o] + EPS_);
    t1[((size_t)b * K2 + o) * QPTS + q] = elu1((acc - rm[o]) * s + bb[o]);
}

// ================= 5) WMMA GEMM + fused bias/BN/ELU epilogue =================
// Y[b, rowOff+m, q] = act( bn( W(MxKc) @ X[b](KcxQ) + bias ) )
// Each wave: one 16-row tile x 64 columns (4 N-tiles), f32 WMMA 16x16x4.
__global__ void gemm_wmma_bn(const float* __restrict__ W, const float* __restrict__ X,
                             float* __restrict__ Y, int M, int Kc,
                             const float* __restrict__ bias,
                             const float* __restrict__ g,  const float* __restrict__ bb,
                             const float* __restrict__ rm, const float* __restrict__ rv,
                             int doElu, int yRowOff, int yBatchStride)
{
    const int gw   = blockIdx.x * 8 + (threadIdx.x >> 5);
    const int lane = threadIdx.x & 31;
    const int numNT = (B_ * QPTS) / 64;
    const int mt = gw / numNT;
    const int nt = gw % numNT;
    if (mt * 16 >= M) return;
    const int m0 = mt * 16;
    const int n0 = nt * 64;
    const int b  = n0 / QPTS;
    const int q0 = n0 % QPTS;
    const int half = lane >> 4;
    const int l16  = lane & 15;
    // A fragment: lanes 0-15 K={0,1}, lanes 16-31 K={2,3} (per ISA 16x4 layout)
    const float* Wrow  = W + (size_t)(m0 + l16) * Kc + 2 * half;
    const float* Xbase = X + (size_t)b * Kc * QPTS + (size_t)(2 * half) * QPTS + q0 + l16;
    v8f acc0 = {}, acc1 = {}, acc2 = {}, acc3 = {};
    for (int kk = 0; kk < Kc; kk += 4) {
        if (kk + 16 < Kc)   // NT prefetch of the X panel one unrolled slab ahead
            __builtin_prefetch(Xbase + (size_t)(kk + 16) * QPTS, 0, 0);
        v2f a;
        a.x = Wrow[kk];
        a.y = Wrow[kk + 1];
        const float* pX = Xbase + (size_t)kk * QPTS;
        v2f b0, b1, b2, b3;
        b0.x = pX[0];  b0.y = pX[QPTS + 0];
        b1.x = pX[16]; b1.y = pX[QPTS + 16];
        b2.x = pX[32]; b2.y = pX[QPTS + 32];
        b3.x = pX[48]; b3.y = pX[QPTS + 48];
        acc0 = __builtin_amdgcn_wmma_f32_16x16x4_f32(false, a, false, b0, (short)0, acc0, false, false);
        acc1 = __builtin_amdgcn_wmma_f32_16x16x4_f32(false, a, false, b1, (short)0, acc1, false, false);
        acc2 = __builtin_amdgcn_wmma_f32_16x16x4_f32(false, a, false, b2, (short)0, acc2, false, false);
        acc3 = __builtin_amdgcn_wmma_f32_16x16x4_f32(false, a, false, b3, (short)0, acc3, false, false);
    }
    // C/D layout: VGPR r, lanes 0-15 -> M=m0+r, lanes 16-31 -> M=m0+r+8; N=l16
    float* Yb = Y + (size_t)b * yBatchStride + (size_t)yRowOff * QPTS + q0 + l16;
    v8f accs[4] = {acc0, acc1, acc2, acc3};
    #pragma unroll
    for (int t = 0; t < 4; ++t) {
        #pragma unroll
        for (int r = 0; r < 8; ++r) {
            int m = m0 + r + 8 * half;
            float s   = g[m] * rsqrtf(rv[m] + EPS_);
            float off = (bias[m] - rm[m]) * s + bb[m];
            float yv  = accs[t][r] * s + off;
            if (doElu) yv = elu1(yv);
            Yb[(size_t)m * QPTS + t * 16] = yv;
        }
    }
}

// ================= 6) fused f1 -> f2 MLP =================
__global__ void f_kernel(const float* __restrict__ phat,
                         const float* __restrict__ w1, const float* __restrict__ b1,
                         const float* __restrict__ g1, const float* __restrict__ be1,
                         const float* __restrict__ rm1, const float* __restrict__ rv1,
                         const float* __restrict__ w2, const float* __restrict__ b2,
                         const float* __restrict__ g2, const float* __restrict__ be2,
                         const float* __restrict__ rm2, const float* __restrict__ rv2,
                         float* __restrict__ f2)
{
    int i = blockIdx.x * 256 + threadIdx.x;      // B*K*Q
    int q = i & (QPTS - 1);
    int k = (i >> 10) & (KNN - 1);
    int b = i >> 14;
    const float* ph = phat + (size_t)b * 3 * KNN * QPTS + (size_t)k * QPTS + q;
    float p0 = ph[0], p1 = ph[(size_t)KNN * QPTS], p2 = ph[(size_t)2 * KNN * QPTS];
    float f1v[MID];
    #pragma unroll
    for (int m = 0; m < MID; ++m) {
        float a = b1[m] + w1[m*3+0]*p0 + w1[m*3+1]*p1 + w1[m*3+2]*p2;
        float s = g1[m] * rsqrtf(rv1[m] + EPS_);
        f1v[m] = elu1((a - rm1[m]) * s + be1[m]);
    }
    #pragma unroll 4
    for (int m = 0; m < MID; ++m) {
        float a = b2[m];
        #pragma unroll
        for (int c = 0; c < MID; ++c) a += w2[m*MID+c] * f1v[c];
        float s = g2[m] * rsqrtf(rv2[m] + EPS_);
        f2[((size_t)b * MID + m) * KNN * QPTS + (size_t)k * QPTS + q] =
            elu1((a - rm2[m]) * s + be2[m]);
    }
}

// ================= 7) fused xh-gather × T contraction + cw1 -> dw =================
__global__ void dw_kernel(const float* __restrict__ f2, const float* __restrict__ x,
                          const int* __restrict__ knn, const float* __restrict__ T,
                          const float* __restrict__ cw1, const float* __restrict__ cb1,
                          float* __restrict__ dw)
{
    int i = blockIdx.x * 256 + threadIdx.x;      // B*CCAT*Q
    int q = i & (QPTS - 1);
    int c = (i >> 10) % CCAT;
    int b = i / (CCAT * QPTS);
    float xv[KNN];
    if (c < MID) {
        const float* f = f2 + ((size_t)b * MID + c) * KNN * QPTS + q;
        #pragma unroll
        for (int j = 0; j < KNN; ++j) xv[j] = f[(size_t)j * QPTS];
    } else {
        const float* xb = x + (size_t)b * CIN * NPTS + (size_t)(c - MID) * NPTS;
        const int* id = knn + (size_t)b * KNN * QPTS + q;
        #pragma unroll
        for (int j = 0; j < KNN; ++j) xv[j] = xb[id[(size_t)j * QPTS]];
    }
    float xt[KNN];
    #pragma unroll
    for (int k = 0; k < KNN; ++k) xt[k] = 0.f;
    const float* Tb = T + (size_t)b * K2 * QPTS + q;
    #pragma unroll 4
    for (int j = 0; j < KNN; ++j) {
        float xj = xv[j];
        #pragma unroll
        for (int k = 0; k < KNN; ++k) xt[k] += xj * Tb[(size_t)(j * 16 + k) * QPTS];
    }
    #pragma unroll
    for (int d = 0; d < DMUL; ++d) {
        int oc = c * DMUL + d;
        float a = cb1[oc];
        #pragma unroll
        for (int k = 0; k < KNN; ++k) a += cw1[(size_t)oc * KNN + k] * xt[k];
        dw[((size_t)b * CCAT * DMUL + oc) * QPTS + q] = a;
    }
}

// ====================================================================
extern "C" void kernel_launch(void* const* d_in, const int* in_sizes, int n_in,
                              void* d_out, int out_size, void* d_ws, size_t ws_size,
                              hipStream_t stream)
{
    (void)in_sizes; (void)n_in; (void)out_size; (void)ws_size;
    const float* x    = (const float*)d_in[0];
    const float* w1   = (const float*)d_in[1];
    const float* b1   = (const float*)d_in[2];
    const float* bn1g = (const float*)d_in[3],  *bn1b = (const float*)d_in[4];
    const float* bn1m = (const float*)d_in[5],  *bn1v = (const float*)d_in[6];
    const float* w2   = (const float*)d_in[7];
    const float* b2   = (const float*)d_in[8];
    const float* bn2g = (const float*)d_in[9],  *bn2b = (const float*)d_in[10];
    const float* bn2m = (const float*)d_in[11], *bn2v = (const float*)d_in[12];
    const float* sw1  = (const float*)d_in[13];
    const float* sb1  = (const float*)d_in[14];
    const float* sbn1g = (const float*)d_in[15], *sbn1b = (const float*)d_in[16];
    const float* sbn1m = (const float*)d_in[17], *sbn1v = (const float*)d_in[18];
    const float* sw2  = (const float*)d_in[19];
    const float* sb2  = (const float*)d_in[20];
    const float* sbn2g = (const float*)d_in[21], *sbn2b = (const float*)d_in[22];
    const float* sbn2m = (const float*)d_in[23], *sbn2v = (const float*)d_in[24];
    const float* sw3  = (const float*)d_in[25];
    const float* sb3  = (const float*)d_in[26];
    const float* sbn3g = (const float*)d_in[27], *sbn3b = (const float*)d_in[28];
    const float* sbn3m = (const float*)d_in[29], *sbn3v = (const float*)d_in[30];
    const float* cw1  = (const float*)d_in[31];
    const float* cb1  = (const float*)d_in[32];
    const float* cw2  = (const float*)d_in[33];
    const float* cb2  = (const float*)d_in[34];
    const float* cbng = (const float*)d_in[35], *cbnb = (const float*)d_in[36];
    const float* cbnm = (const float*)d_in[37], *cbnv = (const float*)d_in[38];

    float* ws   = (float*)d_ws;
    int*   qi   = (int*)(ws + OFF_QI);
    int*   knn  = (int*)(ws + OFF_KNN);
    float* phat = ws + OFF_PHAT;
    float* t1   = ws + OFF_T1;
    float* t2   = ws + OFF_T2;
    float* TT   = ws + OFF_TT;
    float* f2   = ws + OFF_F2;
    float* dwb  = ws + OFF_DW;
    float* out  = (float*)d_out;

    // 1) FPS: 1 block/batch, dists in LDS (64KB + 8KB reduction)
    fps_kernel<<<B_, 1024, (NPTS + 2048) * sizeof(float), stream>>>(x, qi);
    // 2) kNN top-32 (TDM double-buffered LDS staging), keep every 2nd
    knn_kernel<<<B_ * (QPTS / 256), 256, 0, stream>>>(x, qi, knn);
    // 3) p_hat + query coords -> d_out rows 0..2
    phat_kernel<<<(B_ * 3 * KNN * QPTS) / 256, 256, 0, stream>>>(x, qi, knn, phat, out);
    // 4) t1
    t1_kernel<<<(B_ * K2 * QPTS) / 256, 256, 0, stream>>>(
        phat, sw1, sb1, sbn1g, sbn1b, sbn1m, sbn1v, t1);
    // 5) t2 = elu(bn(sw2 @ t1 + sb2))   [WMMA]
    gemm_wmma_bn<<<(16 * 128) / 8, 256, 0, stream>>>(
        sw2, t1, t2, K2, K2, sb2, sbn2g, sbn2b, sbn2m, sbn2v, 1, 0, K2 * QPTS);
    // 6) T = bn(sw3 @ t2 + sb3)         [WMMA]
    gemm_wmma_bn<<<(16 * 128) / 8, 256, 0, stream>>>(
        sw3, t2, TT, K2, K2, sb3, sbn3g, sbn3b, sbn3m, sbn3v, 0, 0, K2 * QPTS);
    // 7) f2 (fused two-layer MLP on p_hat)
    f_kernel<<<(B_ * KNN * QPTS) / 256, 256, 0, stream>>>(
        phat, w1, b1, bn1g, bn1b, bn1m, bn1v,
        w2, b2, bn2g, bn2b, bn2m, bn2v, f2);
    // 8) dw (fused [f2;gather(x)] x T, then cw1 + cb1)
    dw_kernel<<<(B_ * CCAT * QPTS) / 256, 256, 0, stream>>>(
        f2, x, knn, TT, cw1, cb1, dwb);
    // 9) out rows 3..130 = elu(bn(cw2 @ dw + cb2))  [WMMA]
    gemm_wmma_bn<<<(8 * 128) / 8, 256, 0, stream>>>(
        cw2, dwb, out, COUT, CCAT * DMUL, cb2, cbng, cbnb, cbnm, cbnv, 1, 3, 131 * QPTS);
}